// DenseCaps_48653389529209
// MI455X (gfx1250) — compile-verified
//
#include <hip/hip_runtime.h>

#define BB 32
#define II 2048
#define OO 64
#define DD 16
#define EPSF 1e-7f
#define ICH 128          // i's per workgroup chunk
#define IB  4            // i's per softmax round
#define NBG 8            // BB/4 b-groups
#define NCH (II/ICH)     // 16 chunks

typedef __attribute__((ext_vector_type(2))) float v2f;
typedef __attribute__((ext_vector_type(8))) float v8f;

// ---------------------------------------------------------------- zero scratch
__global__ void dc_zero(float* __restrict__ p, int n) {
    int t = blockIdx.x * blockDim.x + threadIdx.x;
    if (t < n) p[t] = 0.0f;
}

// ------------------------------------------------- fused (E-step +) M-step pass
// Workgroup: 256 threads (8 waves). blockIdx.x = b-group (4 batches),
// blockIdx.y = i-chunk. Wave w owns col-tiles 2w,2w+1 (8 output caps).
// Per i: one WMMA F32 16x16x4 per col-tile computes votes for
// rows (b_local*4+p) x cols (o_local*4+r). Prev-iteration mu/invsigma/base
// are hoisted into registers (their lane indices are i-invariant). E-step
// softmax over O is done in LDS, batched IB capsules per barrier round.
__global__ __launch_bounds__(256) void dc_mstep(
    const float* __restrict__ poses,   // [B,I,4,4]
    const float* __restrict__ ain,     // [B,I]
    const float* __restrict__ w,       // [O,I,4,4]
    float* __restrict__ S1,            // [B,O,D] partial sum r*V
    float* __restrict__ S2,            // [B,O,D] partial sum r*V^2
    float* __restrict__ RS,            // [B,O]   partial sum r
    const float* __restrict__ MU,      // [B,O,D] prev mu
    const float* __restrict__ ISG,     // [B,O,D] prev 1/sigma2
    const float* __restrict__ BASE,    // [B,O]   log(a+eps)-0.5*sum log sigma2
    int pass)
{
    __shared__ float lds_ain [4*ICH];
    __shared__ float lds_acc [IB*4*OO];   // [ib][bl][o]
    __shared__ float lds_R   [IB*4*OO];
    __shared__ float lds_redm[IB*8];      // [ib][wave]
    __shared__ float lds_reds[IB*8];

    const int tid  = threadIdx.x;
    const int bg   = blockIdx.x;      // 0..7
    const int i0   = blockIdx.y * ICH;
    const int lane = tid & 31;
    const int wave = tid >> 5;        // 0..7
    const int half = lane >> 4;       // 0/1 selects K pair for WMMA operands
    const int lh   = lane & 15;

    for (int k = tid; k < 4*ICH; k += 256) {
        int bl = k / ICH, ii = k % ICH;
        lds_ain[k] = ain[(size_t)(bg*4 + bl)*II + i0 + ii];
    }
    __syncthreads();

    // static per-lane matrix coordinates
    const int b_l = lh >> 2, p  = lh & 3;   // A rows:  M = b_l*4 + p
    const int o_l = lh >> 2, rr = lh & 3;   // B cols:  N = o_l*4 + r
    const int q0  = half * 2;               // K pair {q0,q0+1}
    const int oA  = wave*8 + o_l;           // col-tile 2*wave
    const int oB  = oA + 4;                 // col-tile 2*wave+1
    const int bb  = bg*4 + b_l;

    const float* Abase  = poses + ((size_t)bb*II)*16 + p*4 + q0;
    const float* BbaseA = w     + ((size_t)oA*II)*16 + q0*4 + rr;
    const float* BbaseB = w     + ((size_t)oB*II)*16 + q0*4 + rr;

    // hoist prev-iteration stats into registers: lane slot (j) -> (b_l,o,d)
    // mapping does not depend on i, so these are loop-invariant.
    float muA[8], muB[8], isgA[8], isgB[8];
    float xbase = 0.0f;
    if (pass) {
        #pragma unroll
        for (int j = 0; j < 8; ++j) {
            int M  = half*8 + j;
            int bl = M >> 2;
            int d  = (M & 3)*4 + rr;
            size_t gA = ((size_t)(bg*4 + bl)*OO + oA)*DD + d;
            size_t gB = ((size_t)(bg*4 + bl)*OO + oB)*DD + d;
            muA[j]  = MU [gA];  muB[j]  = MU [gB];
            isgA[j] = ISG[gA];  isgB[j] = ISG[gB];
        }
        xbase = BASE[bg*4*OO + tid];   // softmax phase: tid = bl*64 + o
    }

    v8f acc1A = {}; v8f acc2A = {}; v8f acc1B = {}; v8f acc2B = {};
    float rsA0 = 0.f, rsA1 = 0.f, rsB0 = 0.f, rsB1 = 0.f;

    for (int ii0 = 0; ii0 < ICH; ii0 += IB) {
        // warm WGP$ for next batch of strided weight gathers
        {
            size_t nofs = (size_t)(i0 + ii0 + IB) * 16;
            __builtin_prefetch(BbaseA + nofs, 0, 1);
            __builtin_prefetch(BbaseB + nofs, 0, 1);
        }
        v8f cA[IB], cB[IB];
        #pragma unroll
        for (int t = 0; t < IB; ++t) {
            const size_t iofs = (size_t)(i0 + ii0 + t) * 16;
            v2f a = *(const v2f*)(Abase + iofs);           // poses pair
            v2f b0; b0.x = BbaseA[iofs]; b0.y = BbaseA[iofs + 4];
            v2f b1; b1.x = BbaseB[iofs]; b1.y = BbaseB[iofs + 4];
            v8f z = {};
            cA[t] = __builtin_amdgcn_wmma_f32_16x16x4_f32(false, a, false, b0,
                                                          (short)0, z, false, false);
            cB[t] = __builtin_amdgcn_wmma_f32_16x16x4_f32(false, a, false, b1,
                                                          (short)0, z, false, false);
        }

        float rvA0[IB], rvA1[IB], rvB0[IB], rvB1[IB];

        if (pass) {
            // ---- E-step partials: sum_d diff^2 * invsigma, per capsule ----
            float sA0[IB], sA1[IB], sB0[IB], sB1[IB];
            #pragma unroll
            for (int t = 0; t < IB; ++t) {
                sA0[t] = sA1[t] = sB0[t] = sB1[t] = 0.f;
                #pragma unroll
                for (int j = 0; j < 8; ++j) {
                    float dA = cA[t][j] - muA[j];
                    float dB = cB[t][j] - muB[j];
                    float tA = dA*dA*isgA[j];
                    float tB = dB*dB*isgB[j];
                    if (j < 4) { sA0[t] += tA; sB0[t] += tB; }
                    else       { sA1[t] += tA; sB1[t] += tB; }
                }
                sA0[t] += __shfl_xor(sA0[t],1,32); sA0[t] += __shfl_xor(sA0[t],2,32);
                sA1[t] += __shfl_xor(sA1[t],1,32); sA1[t] += __shfl_xor(sA1[t],2,32);
                sB0[t] += __shfl_xor(sB0[t],1,32); sB0[t] += __shfl_xor(sB0[t],2,32);
                sB1[t] += __shfl_xor(sB1[t],1,32); sB1[t] += __shfl_xor(sB1[t],2,32);
            }
            #pragma unroll
            for (int t = 0; t < IB; ++t) lds_acc[t*4*OO + tid] = 0.f;
            __syncthreads();
            if ((lane & 3) == 0) {
                #pragma unroll
                for (int t = 0; t < IB; ++t) {
                    atomicAdd(&lds_acc[t*4*OO + (half*2    )*OO + oA], sA0[t]);
                    atomicAdd(&lds_acc[t*4*OO + (half*2 + 1)*OO + oA], sA1[t]);
                    atomicAdd(&lds_acc[t*4*OO + (half*2    )*OO + oB], sB0[t]);
                    atomicAdd(&lds_acc[t*4*OO + (half*2 + 1)*OO + oB], sB1[t]);
                }
            }
            __syncthreads();

            // ---- softmax over O within 64-thread groups, IB at a time ----
            float xv[IB], ev[IB];
            #pragma unroll
            for (int t = 0; t < IB; ++t) {
                float x = xbase - 0.5f * lds_acc[t*4*OO + tid];
                xv[t] = x;
                float m = x;
                m = fmaxf(m, __shfl_xor(m, 1,32));
                m = fmaxf(m, __shfl_xor(m, 2,32));
                m = fmaxf(m, __shfl_xor(m, 4,32));
                m = fmaxf(m, __shfl_xor(m, 8,32));
                m = fmaxf(m, __shfl_xor(m,16,32));
                if (lane == 0) lds_redm[t*8 + wave] = m;
            }
            __syncthreads();
            #pragma unroll
            for (int t = 0; t < IB; ++t) {
                float m = fmaxf(lds_redm[t*8 + (wave & ~1)],
                                lds_redm[t*8 + (wave |  1)]);
                float e = expf(xv[t] - m);
                ev[t] = e;
                float s = e;
                s += __shfl_xor(s, 1,32);
                s += __shfl_xor(s, 2,32);
                s += __shfl_xor(s, 4,32);
                s += __shfl_xor(s, 8,32);
                s += __shfl_xor(s,16,32);
                if (lane == 0) lds_reds[t*8 + wave] = s;
            }
            __syncthreads();
            #pragma unroll
            for (int t = 0; t < IB; ++t) {
                float s = lds_reds[t*8 + (wave & ~1)] +
                          lds_reds[t*8 + (wave |  1)];
                lds_R[t*4*OO + tid] = ev[t] / s;
            }
            __syncthreads();
            #pragma unroll
            for (int t = 0; t < IB; ++t) {
                float ai0 = lds_ain[(half*2    )*ICH + ii0 + t];
                float ai1 = lds_ain[(half*2 + 1)*ICH + ii0 + t];
                rvA0[t] = lds_R[t*4*OO + (half*2    )*OO + oA] * ai0;
                rvA1[t] = lds_R[t*4*OO + (half*2 + 1)*OO + oA] * ai1;
                rvB0[t] = lds_R[t*4*OO + (half*2    )*OO + oB] * ai0;
                rvB1[t] = lds_R[t*4*OO + (half*2 + 1)*OO + oB] * ai1;
            }
        } else {
            // first iteration: uniform R = 1/O
            #pragma unroll
            for (int t = 0; t < IB; ++t) {
                float ai0 = lds_ain[(half*2    )*ICH + ii0 + t];
                float ai1 = lds_ain[(half*2 + 1)*ICH + ii0 + t];
                rvA0[t] = ai0 * (1.0f/OO); rvA1[t] = ai1 * (1.0f/OO);
                rvB0[t] = rvA0[t];         rvB1[t] = rvA1[t];
            }
        }

        // ---- M-step register accumulation ----
        #pragma unroll
        for (int t = 0; t < IB; ++t) {
            #pragma unroll
            for (int j = 0; j < 8; ++j) {
                float rA = (j < 4) ? rvA0[t] : rvA1[t];
                float rB = (j < 4) ? rvB0[t] : rvB1[t];
                acc1A[j] += rA * cA[t][j];
                acc2A[j] += rA * cA[t][j] * cA[t][j];
                acc1B[j] += rB * cB[t][j];
                acc2B[j] += rB * cB[t][j] * cB[t][j];
            }
            rsA0 += rvA0[t]; rsA1 += rvA1[t];
            rsB0 += rvB0[t]; rsB1 += rvB1[t];
        }
    }

    // flush partials (each (b,o,d) owned by exactly one lane-register here;
    // atomics only resolve cross-workgroup i-chunk duplication, L2-resident)
    #pragma unroll
    for (int j = 0; j < 8; ++j) {
        int M = half*8 + j;
        int b = bg*4 + (M >> 2);
        int d = (M & 3)*4 + rr;
        atomicAdd(&S1[((size_t)b*OO + oA)*DD + d], acc1A[j]);
        atomicAdd(&S2[((size_t)b*OO + oA)*DD + d], acc2A[j]);
        atomicAdd(&S1[((size_t)b*OO + oB)*DD + d], acc1B[j]);
        atomicAdd(&S2[((size_t)b*OO + oB)*DD + d], acc2B[j]);
    }
    if ((lane & 3) == 0) {
        int bF = bg*4 + half*2, bS = bF + 1;
        atomicAdd(&RS[bF*OO + oA], rsA0);
        atomicAdd(&RS[bS*OO + oA], rsA1);
        atomicAdd(&RS[bF*OO + oB], rsB0);
        atomicAdd(&RS[bS*OO + oB], rsB1);
    }
}

// ------------------------------------------------------------- finalize a pass
__global__ __launch_bounds__(256) void dc_finalize(
    const float* __restrict__ S1, const float* __restrict__ S2,
    const float* __restrict__ RS,
    const float* __restrict__ beta_a, const float* __restrict__ beta_u,
    float* __restrict__ MU, float* __restrict__ ISG, float* __restrict__ BASE,
    float inv_temp, int final_pass,
    float* __restrict__ out_mu, float* __restrict__ out_a)
{
    int t = blockIdx.x*256 + threadIdx.x;       // (b*O + o)
    if (t >= BB*OO) return;
    int o = t & (OO - 1);
    float rsum = RS[t] + EPSF;
    float inv  = 1.0f / rsum;
    float mu[DD], isg[DD];
    float sumlog = 0.0f;
    #pragma unroll
    for (int d = 0; d < DD; ++d) {
        float m  = S1[(size_t)t*DD + d] * inv;
        float s2 = S2[(size_t)t*DD + d] * inv - m*m + EPSF;
        s2 = fmaxf(s2, EPSF);
        mu[d]  = m;
        isg[d] = 1.0f / s2;
        sumlog += logf(s2);
    }
    float cost = rsum * ((float)DD * beta_u[o] + 0.5f * sumlog);
    float zv   = inv_temp * (beta_a[o] - cost);
    float av   = 1.0f / (1.0f + expf(-zv));
    if (final_pass) {
        #pragma unroll
        for (int d = 0; d < DD; ++d) out_mu[(size_t)t*DD + d] = mu[d];
        out_a[t] = av;
    } else {
        #pragma unroll
        for (int d = 0; d < DD; ++d) {
            MU [(size_t)t*DD + d] = mu[d];
            ISG[(size_t)t*DD + d] = isg[d];
        }
        BASE[t] = logf(av + EPSF) - 0.5f * sumlog;
    }
}

extern "C" void kernel_launch(void* const* d_in, const int* in_sizes, int n_in,
                              void* d_out, int out_size, void* d_ws, size_t ws_size,
                              hipStream_t stream) {
    (void)in_sizes; (void)n_in; (void)out_size; (void)ws_size;
    const float* poses  = (const float*)d_in[0];   // [B,I,4,4]
    const float* ain    = (const float*)d_in[1];   // [B,I]
    const float* w      = (const float*)d_in[2];   // [O,I,4,4]
    const float* beta_a = (const float*)d_in[3];   // [1,O]
    const float* beta_u = (const float*)d_in[4];   // [1,O]
    float* out = (float*)d_out;                    // [B,O,4,4] ++ [B,O]
    float* ws  = (float*)d_ws;

    float* S1   = ws;                  // B*O*D
    float* S2   = S1 + BB*OO*DD;       // B*O*D
    float* RS   = S2 + BB*OO*DD;       // B*O
    float* MU   = RS + BB*OO;          // B*O*D
    float* ISG  = MU + BB*OO*DD;       // B*O*D
    float* BASE = ISG + BB*OO*DD;      // B*O

    const int NZ = 2*BB*OO*DD + BB*OO; // S1,S2,RS contiguous
    dim3 mgrid(NBG, NCH);
    for (int pass = 0; pass < 3; ++pass) {
        dc_zero<<<(NZ + 255)/256, 256, 0, stream>>>(S1, NZ);
        dc_mstep<<<mgrid, 256, 0, stream>>>(poses, ain, w, S1, S2, RS,
                                            MU, ISG, BASE, pass);
        dc_finalize<<<(BB*OO + 255)/256, 256, 0, stream>>>(
            S1, S2, RS, beta_a, beta_u, MU, ISG, BASE,
            (float)(1 + pass), (pass == 2) ? 1 : 0, out, out + BB*OO*DD);
    }
}